// VectorQuantizer_85220741087886
// MI455X (gfx1250) — compile-verified
//
#include <hip/hip_runtime.h>
#include <stdint.h>

typedef __attribute__((ext_vector_type(2))) float v2f;
typedef __attribute__((ext_vector_type(8))) float v8f;

#define DIM      256
#define NEMBED   1024
#define NVEC     65536           // 16*64*64
#define NELEM    16777216        // 16*256*64*64
#define ASTRIDE  260             // padded LDS row stride (floats): bank-conflict-free frags
#define KSTEPS   64              // 256 / 4
#define NGROUP   16              // 16 iterations x 4 code-tiles (64 codes) each

// ---------------- Kernel 0: pack codebook into WMMA-B fragment order, half-norms, zero counts ----
// Bpack[((t*64+s)*32 + l)*2 + j] = E[t*16 + (l&15)][s*4 + (l>>4)*2 + j]
__global__ __launch_bounds__(256) void vq_pack_kernel(
    const float* __restrict__ ew, float* __restrict__ Bpack,
    float* __restrict__ enormh, int* __restrict__ counts)
{
    __shared__ float red[256];
    const int c = blockIdx.x;      // code 0..1023
    const int k = threadIdx.x;     // dim  0..255
    float v = ew[c * DIM + k];
    int t = c >> 4;
    int s = k >> 2;
    int l = (((k & 3) >> 1) << 4) + (c & 15);
    int j = k & 1;
    Bpack[(((t << 6) + s) * 32 + l) * 2 + j] = v;
    red[k] = v * v;
    __syncthreads();
    for (int st = 128; st > 0; st >>= 1) {
        if (k < st) red[k] += red[k + st];
        __syncthreads();
    }
    if (k == 0) {
        enormh[c] = 0.5f * red[0];
        counts[c] = 0;
    }
}

// async copy of one 16-byte chunk: global -> LDS (ASYNCcnt-tracked DMA-style copy)
__device__ __forceinline__ void async_b128(uint32_t lds_off, const float* gsrc) {
    asm volatile("global_load_async_to_lds_b128 %0, %1, off"
                 :: "v"(lds_off), "v"(gsrc) : "memory");
}
__device__ __forceinline__ void wait_async0() {
    asm volatile("s_wait_asynccnt 0x0" ::: "memory");
}

// ---------------- Kernel 1: WMMA f32 GEMM + running argmin over 1024 codes ---------------------
// Block = 256 threads (8 wave32), 128 rows (two (b,h) pixel rows). Grid = 512.
// Each wave owns 16 rows; per iteration it multiplies its A-frag against 4 code tiles
// (4 independent accumulator chains), B staged in LDS by async global->LDS copies.
__global__ __launch_bounds__(256) void vq_argmin_kernel(
    const float* __restrict__ z, const float* __restrict__ Bpack,
    const float* __restrict__ enormh, float* __restrict__ idxf,
    int* __restrict__ idxi, int* __restrict__ counts)
{
    extern __shared__ float smem[];
    float* sA = smem;                      // 128 * ASTRIDE floats            (133,120 B)
    float* sB = smem + 128 * ASTRIDE;      // 2 * 16384 floats double buffer  (131,072 B)
    const uint32_t sBbase = (uint32_t)(uintptr_t)sB;   // LDS byte offset (low 32 bits of flat)

    const int tid  = threadIdx.x;
    const int lane = tid & 31;
    const int wv   = tid >> 5;
    const int n0   = blockIdx.x * 128;     // first flattened row
    const int b    = n0 >> 12;
    const int h0   = (n0 >> 6) & 63;

    // ---- stage A tile: rows r = hh*64 + w ; zf[n][c] = z[((b*256+c)*64 + h)*64 + w] ----
    {
        const int w  = tid & 63;
        const int cg = tid >> 6;
        for (int hh = 0; hh < 2; ++hh) {
            const float* zsrc = z + (b << 20) + ((h0 + hh) << 6) + w;
            float* dst = sA + ((hh << 6) + w) * ASTRIDE;
            for (int c = cg; c < DIM; c += 4)
                dst[c] = zsrc[c << 12];
        }
    }

    // issue async B copy for group 0 (64 KB: 16 x b128 per thread)
    {
        const float* src = Bpack + (tid << 2);
        const uint32_t dst = sBbase + (tid << 4);
#pragma unroll
        for (int q = 0; q < 16; ++q)
            async_b128(dst + (q << 12), src + (q << 10));
    }

    float mval[8];
    int   midx[8];
#pragma unroll
    for (int v = 0; v < 8; ++v) { mval[v] = 3.4e38f; midx[v] = 0; }

    // A fragment: lane L holds row M = L&15, K pair (L>>4)*2 within each 4-wide K step
    const float* Abase = sA + ((wv << 4) + (lane & 15)) * ASTRIDE + ((lane >> 4) << 1);
    const int colLane = lane & 15;

    for (int tp = 0; tp < NGROUP; ++tp) {
        wait_async0();         // this wave's async B copies landed in LDS
        __syncthreads();       // everyone's landed; prior buffer reads done

        if (tp + 1 < NGROUP) { // async-prefetch next 4-tile group into other buffer
            const float* src = Bpack + ((tp + 1) << 14) + (tid << 2);
            const uint32_t dst = sBbase + (((tp + 1) & 1) << 16) + (tid << 4);
#pragma unroll
            for (int q = 0; q < 16; ++q)
                async_b128(dst + (q << 12), src + (q << 10));
        }

        const float* B0 = sB + ((tp & 1) << 14) + (lane << 1);
        v8f acc0 = {}, acc1 = {}, acc2 = {}, acc3 = {};
#pragma unroll
        for (int s = 0; s < KSTEPS; ++s) {
            v2f a  = *(const v2f*)(Abase + (s << 2));
            v2f b0 = *(const v2f*)(B0 + (s << 6));
            v2f b1 = *(const v2f*)(B0 + 4096 + (s << 6));
            v2f b2 = *(const v2f*)(B0 + 8192 + (s << 6));
            v2f b3 = *(const v2f*)(B0 + 12288 + (s << 6));
            acc0 = __builtin_amdgcn_wmma_f32_16x16x4_f32(false, a, false, b0, (short)0, acc0, false, false);
            acc1 = __builtin_amdgcn_wmma_f32_16x16x4_f32(false, a, false, b1, (short)0, acc1, false, false);
            acc2 = __builtin_amdgcn_wmma_f32_16x16x4_f32(false, a, false, b2, (short)0, acc2, false, false);
            acc3 = __builtin_amdgcn_wmma_f32_16x16x4_f32(false, a, false, b3, (short)0, acc3, false, false);
        }

        // score = 0.5*||e||^2 - z.e  (monotone in squared distance); codes ascend -> first-min kept
        const int tb = tp << 6;
#define VQ_UPDATE(ACC, Q)                                                    \
        {                                                                    \
            const int code = tb + ((Q) << 4) + colLane;                      \
            const float en = enormh[code];                                   \
            _Pragma("unroll")                                                \
            for (int v = 0; v < 8; ++v) {                                    \
                float sc = en - ACC[v];                                      \
                if (sc < mval[v]) { mval[v] = sc; midx[v] = code; }          \
            }                                                                \
        }
        VQ_UPDATE(acc0, 0)
        VQ_UPDATE(acc1, 1)
        VQ_UPDATE(acc2, 2)
        VQ_UPDATE(acc3, 3)
#undef VQ_UPDATE
    }

    // reduce across the 16-lane halves (lanes 0-15 hold rows v, lanes 16-31 rows v+8)
    const int rowb = n0 + (wv << 4);
#pragma unroll
    for (int v = 0; v < 8; ++v) {
        float mv = mval[v];
        int   mi = midx[v];
#pragma unroll
        for (int off = 8; off >= 1; off >>= 1) {
            float ov = __shfl_xor(mv, off, 32);
            int   oi = __shfl_xor(mi, off, 32);
            if (ov < mv || (ov == mv && oi < mi)) { mv = ov; mi = oi; }
        }
        if ((lane & 15) == 0) {
            int row = rowb + v + ((lane & 16) ? 8 : 0);
            idxf[row] = (float)mi;
            idxi[row] = mi;
            atomicAdd(&counts[mi], 1);
        }
    }
}

// ---------------- Kernel 2: z_q gather + per-block SSE partial sums ---------------------------
__global__ __launch_bounds__(256) void vq_gather_kernel(
    const float* __restrict__ z, const float* __restrict__ ew,
    const int* __restrict__ idxi, float* __restrict__ zq,
    float* __restrict__ partials)
{
    __shared__ int   sIdx[64];
    __shared__ float red[256];
    const int p   = blockIdx.x;     // (b,h) pair 0..1023
    const int tid = threadIdx.x;
    const int b   = p >> 6;
    const int h   = p & 63;
    if (tid < 64) sIdx[tid] = idxi[(p << 6) + tid];
    __syncthreads();

    const int w  = tid & 63;
    const int cg = tid >> 6;
    const int er = sIdx[w] * DIM;
    const int hw = (h << 6) + w;
    float sse = 0.0f;
    for (int c = cg; c < DIM; c += 4) {
        int off = (((b << 8) + c) << 12) + hw;
        float e  = ew[er + c];
        float zv = z[off];
        float d  = e - zv;
        sse += d * d;
        zq[off] = e;
    }
    red[tid] = sse;
    __syncthreads();
    for (int st = 128; st > 0; st >>= 1) {
        if (tid < st) red[tid] += red[tid + st];
        __syncthreads();
    }
    if (tid == 0) partials[p] = red[0];
}

// ---------------- Kernel 3: finalize loss + perplexity ---------------------------------------
__global__ __launch_bounds__(256) void vq_final_kernel(
    const float* __restrict__ partials, const int* __restrict__ counts,
    float* __restrict__ loss_out, float* __restrict__ perp_out)
{
    __shared__ float red[256];
    const int tid = threadIdx.x;

    float s = 0.0f;
    for (int i = tid; i < 1024; i += 256) s += partials[i];
    red[tid] = s;
    __syncthreads();
    for (int st = 128; st > 0; st >>= 1) {
        if (tid < st) red[tid] += red[tid + st];
        __syncthreads();
    }
    if (tid == 0) loss_out[0] = 1.25f * red[0] * (1.0f / (float)NELEM);
    __syncthreads();

    float ap = 0.0f;
    for (int e = tid; e < NEMBED; e += 256) {
        float em = (float)counts[e] * (1.0f / (float)NVEC);
        ap += em * logf(em + 1e-10f);
    }
    red[tid] = ap;
    __syncthreads();
    for (int st = 128; st > 0; st >>= 1) {
        if (tid < st) red[tid] += red[tid + st];
        __syncthreads();
    }
    if (tid == 0) perp_out[0] = expf(-red[0]);
}

extern "C" void kernel_launch(void* const* d_in, const int* in_sizes, int n_in,
                              void* d_out, int out_size, void* d_ws, size_t ws_size,
                              hipStream_t stream) {
    const float* z  = (const float*)d_in[0];   // [16,256,64,64]
    const float* ew = (const float*)d_in[1];   // [1024,256]
    float* out = (float*)d_out;
    char*  ws  = (char*)d_ws;

    float* Bpack    = (float*)(ws);                 // 1,048,576 B
    float* enormh   = (float*)(ws + 1048576);       //     4,096 B
    int*   counts   = (int*)  (ws + 1052672);       //     4,096 B
    int*   idxi     = (int*)  (ws + 1056768);       //   262,144 B
    float* partials = (float*)(ws + 1318912);       //     4,096 B

    float* zq   = out + 1;                          // [16,256,64,64]
    float* idxf = out + 1 + NELEM;                  // [65536,1]
    float* perp = out + 1 + NELEM + NVEC;           // scalar

    vq_pack_kernel<<<NEMBED, 256, 0, stream>>>(ew, Bpack, enormh, counts);

    const size_t smem = (size_t)(128 * ASTRIDE + 2 * 16384) * sizeof(float); // 264,192 B
    vq_argmin_kernel<<<NVEC / 128, 256, smem, stream>>>(z, Bpack, enormh, idxf, idxi, counts);

    vq_gather_kernel<<<1024, 256, 0, stream>>>(z, ew, idxi, zq, partials);

    vq_final_kernel<<<1, 256, 0, stream>>>(partials, counts, out, perp);
}